// LieRE_83150566851036
// MI455X (gfx1250) — compile-verified
//
#include <hip/hip_runtime.h>
#include <hip/hip_bf16.h>

// LieRE rotary embedding on MI455X (gfx1250, wave32, WMMA).
// Phase 1: build skew generator A (64x64) from packed strict-lower entries.
// Phase 2: one workgroup per t: R = expm((t-offset)*A) via scaling-&-squaring
//          with Horner Taylor (all 64x64 matmuls via V_WMMA_F32_16X16X4_F32 in
//          LDS), then out[bh,t,:] = x[bh,t,:] @ R for all 256 bh rows.

#define DD   64
#define TT   2048
#define BH   256          // B*H
#define LDR  65           // LDS row stride (floats) -> conflict-free columns
#define KT   10           // Taylor terms (||B||<=0.25 -> err ~1e-14)

typedef float v2f __attribute__((ext_vector_type(2)));
typedef float v8f __attribute__((ext_vector_type(8)));

// ---------------------------------------------------------------- Phase 1
__global__ void liere_build_A(const float* __restrict__ entries,
                              float* __restrict__ A) {
    int idx = blockIdx.x * blockDim.x + threadIdx.x;   // 0..4095
    if (idx >= DD * DD) return;
    int r = idx >> 6, c = idx & 63;
    float v = 0.0f;
    if (r > c)      v =  entries[r * (r - 1) / 2 + c];
    else if (r < c) v = -entries[c * (c - 1) / 2 + r];
    A[idx] = v;
}

// ------------------------------------------------- 64x64 LDS matmul (WMMA)
// dst = scale * (a @ b) (+ I if addIdent). dst must not alias a or b.
// 8 waves: wave w -> tileM = w&3, tileN in {2*(w>>2), 2*(w>>2)+1}.
__device__ __forceinline__ void mm64_lds(float* __restrict__ dst,
                                         const float* __restrict__ a,
                                         const float* __restrict__ b,
                                         float scale, int addIdent) {
    const int lane = threadIdx.x & 31;
    const int wave = threadIdx.x >> 5;
    const int lo   = lane & 15;
    const int hi   = lane >> 4;
    const int tm   = wave & 3;
    const int tn0  = (wave >> 2) * 2;

    v8f c0 = {}; v8f c1 = {};
    #pragma unroll
    for (int kk = 0; kk < 16; ++kk) {
        const int k0 = kk * 4 + 2 * hi;            // A/B K layout (ISA 7.12.2)
        v2f av, b0v, b1v;
        av.x  = a[(tm * 16 + lo) * LDR + k0];
        av.y  = a[(tm * 16 + lo) * LDR + k0 + 1];
        b0v.x = b[(k0    ) * LDR + tn0 * 16 + lo];
        b0v.y = b[(k0 + 1) * LDR + tn0 * 16 + lo];
        b1v.x = b[(k0    ) * LDR + (tn0 + 1) * 16 + lo];
        b1v.y = b[(k0 + 1) * LDR + (tn0 + 1) * 16 + lo];
        c0 = __builtin_amdgcn_wmma_f32_16x16x4_f32(false, av, false, b0v,
                                                   (short)0, c0, false, false);
        c1 = __builtin_amdgcn_wmma_f32_16x16x4_f32(false, av, false, b1v,
                                                   (short)0, c1, false, false);
    }
    #pragma unroll
    for (int j = 0; j < 8; ++j) {
        const int r  = tm * 16 + j + 8 * hi;       // C/D: VGPR j -> rows j, j+8
        float v0 = c0[j] * scale;
        float v1 = c1[j] * scale;
        if (addIdent) {
            if (r == tn0 * 16 + lo)       v0 += 1.0f;
            if (r == (tn0 + 1) * 16 + lo) v1 += 1.0f;
        }
        dst[r * LDR + tn0 * 16 + lo]       = v0;
        dst[r * LDR + (tn0 + 1) * 16 + lo] = v1;
    }
    __syncthreads();
}

// ---------------------------------------------------------------- Phase 2
__global__ void __launch_bounds__(256)
liere_rotate(const float* __restrict__ x, const float* __restrict__ Ag,
             const int* __restrict__ offset, float* __restrict__ out) {
    __shared__ float sBm[DD * LDR];      // scaled generator, later x tile
    __shared__ float sE[2][DD * LDR];    // expm ping-pong
    __shared__ float sRow[DD];
    __shared__ float sFac;
    __shared__ int   sS;

    const int th = threadIdx.x;
    const int t  = blockIdx.x;
    const float tv = (float)(t - offset[0]);

    // load raw A into sBm
    for (int idx = th; idx < DD * DD; idx += 256)
        sBm[(idx >> 6) * LDR + (idx & 63)] = Ag[idx];
    __syncthreads();

    // inf-norm bound -> squaring count s, scale factor tv/2^s
    if (th < DD) {
        float acc = 0.0f;
        for (int c = 0; c < DD; ++c) acc += fabsf(sBm[th * LDR + c]);
        sRow[th] = acc;
    }
    __syncthreads();
    if (th == 0) {
        float nrm = 0.0f;
        for (int r = 0; r < DD; ++r) nrm = fmaxf(nrm, sRow[r]);
        float m = nrm * fabsf(tv);
        int s = 0;
        if (m > 0.25f) {
            s = (int)ceilf(log2f(m * 4.0f));
            if (s > 30) s = 30;
        }
        sS   = s;
        sFac = tv * exp2f((float)-s);
    }
    __syncthreads();
    const float fac = sFac;
    const int   s   = sS;

    // Bm = (tv/2^s) * A  (in place)
    for (int idx = th; idx < DD * DD; idx += 256)
        sBm[(idx >> 6) * LDR + (idx & 63)] *= fac;

    // Horner Taylor: E = I + Bm/KT; E = I + (Bm/k) @ E, k = KT-1..1
    for (int idx = th; idx < DD * DD; idx += 256) {
        int r = idx >> 6, c = idx & 63;
        sE[0][r * LDR + c] = sBm[r * LDR + c] * (1.0f / KT) + (r == c ? 1.0f : 0.0f);
    }
    __syncthreads();

    int cur = 0;
    for (int k = KT - 1; k >= 1; --k) {
        mm64_lds(sE[cur ^ 1], sBm, sE[cur], 1.0f / (float)k, 1);
        cur ^= 1;
    }
    // squarings: R = E^(2^s)
    for (int i = 0; i < s; ++i) {
        mm64_lds(sE[cur ^ 1], sE[cur], sE[cur], 1.0f, 0);
        cur ^= 1;
    }
    const float* R = sE[cur];

    // ------- apply: out[bh,t,:] = x[bh,t,:] @ R, 4 blocks of 64 bh-rows ----
    const int lane = th & 31, wave = th >> 5;
    const int lo = lane & 15, hi = lane >> 4;
    const int tm = wave & 3, tn0 = (wave >> 2) * 2;
    const int xr = th >> 2, xq = th & 3;           // row / 16-col quarter

    for (int blk = 0; blk < 4; ++blk) {
        const size_t base =
            ((size_t)(blk * 64 + xr) * TT + (size_t)t) * DD + (size_t)xq * 16;
        const float4* __restrict__ xs = (const float4*)(x + base);
        float4 f0 = xs[0], f1 = xs[1], f2 = xs[2], f3 = xs[3];
        if (blk < 3)   // warm L2 for the next bh block
            __builtin_prefetch(x + base + (size_t)64 * TT * DD, 0, 1);
        float* xt = sBm;                           // reuse generator buffer
        float4* xd = (float4*)(xt + xr * LDR + xq * 16);
        xd[0] = f0; xd[1] = f1; xd[2] = f2; xd[3] = f3;
        __syncthreads();

        v8f c0 = {}; v8f c1 = {};
        #pragma unroll
        for (int kk = 0; kk < 16; ++kk) {
            const int k0 = kk * 4 + 2 * hi;
            v2f av, b0v, b1v;
            av.x  = xt[(tm * 16 + lo) * LDR + k0];
            av.y  = xt[(tm * 16 + lo) * LDR + k0 + 1];
            b0v.x = R[(k0    ) * LDR + tn0 * 16 + lo];
            b0v.y = R[(k0 + 1) * LDR + tn0 * 16 + lo];
            b1v.x = R[(k0    ) * LDR + (tn0 + 1) * 16 + lo];
            b1v.y = R[(k0 + 1) * LDR + (tn0 + 1) * 16 + lo];
            c0 = __builtin_amdgcn_wmma_f32_16x16x4_f32(false, av, false, b0v,
                                                       (short)0, c0, false, false);
            c1 = __builtin_amdgcn_wmma_f32_16x16x4_f32(false, av, false, b1v,
                                                       (short)0, c1, false, false);
        }
        #pragma unroll
        for (int j = 0; j < 8; ++j) {
            const int r   = tm * 16 + j + 8 * hi;
            const size_t ob =
                ((size_t)(blk * 64 + r) * TT + (size_t)t) * DD;
            out[ob + tn0 * 16 + lo]       = c0[j];
            out[ob + (tn0 + 1) * 16 + lo] = c1[j];
        }
        __syncthreads();                           // before xt reuse
    }
}

// ---------------------------------------------------------------- launcher
extern "C" void kernel_launch(void* const* d_in, const int* in_sizes, int n_in,
                              void* d_out, int out_size, void* d_ws, size_t ws_size,
                              hipStream_t stream) {
    const float* x       = (const float*)d_in[0];
    const float* entries = (const float*)d_in[1];
    const int*   offset  = (const int*)d_in[2];
    float*       out     = (float*)d_out;
    float*       A       = (float*)d_ws;          // 64*64 f32 = 16 KB

    liere_build_A<<<(DD * DD + 255) / 256, 256, 0, stream>>>(entries, A);
    liere_rotate<<<TT, 256, 0, stream>>>(x, A, offset, out);
}